// DifferentiableAmbiguity_14053132993030
// MI455X (gfx1250) — compile-verified
//
#include <hip/hip_runtime.h>

// CDNA5 (gfx1250) wave32 WMMA types
typedef __attribute__((ext_vector_type(2))) float v2f;   // f32 16x16x4 A/B operand: 2 VGPRs
typedef __attribute__((ext_vector_type(8))) float v8f;   // f32 16x16 C/D accumulator: 8 VGPRs

#define NFFT   1024
#define NMASK  1023
#define NBATCH 16
#define NKT    33          // lag row-tiles per batch: k = 0..512 (symmetry folds the rest)

// ---------------------------------------------------------------------------
// Kernel 0: zero the per-batch max slots in workspace (harness poisons ws with
// 0xAA once and never re-poisons, so we must re-init every call).
// ---------------------------------------------------------------------------
__global__ void amb_init_max(unsigned int* __restrict__ bmax) {
    if (threadIdx.x < NBATCH) bmax[threadIdx.x] = 0u;
}

// ---------------------------------------------------------------------------
// Kernel 1: ambiguity surface via dense DFT expressed as f32 WMMA GEMM,
// exploiting central symmetry chi[(-k)%N, (-j)%N] == chi[k,j]
// (since X[N-k,N-j] = e^{-2pi i jk/N} conj(X[k,j])): only lags k=0..512 are
// computed (33 row-tiles/batch instead of 64 -> 1.94x fewer WMMAs), and each
// k in [1,511] result is mirrored to ((-k)%N, (-j)%N). Every output cell is
// written exactly once -> deterministic.
//
//   grid.x = 16 batches * 33 lag-tiles = 528
//   grid.y = 16 column super-tiles (64 freqs each; 4 waves * 16 freqs)
//   block  = 128 threads = 4 wave32
// Each wave owns one 16(lag) x 16(freq) f32 tile, K-loop over m=0..1023 in
// steps of 4 using V_WMMA_F32_16X16X4_F32 (full f32 precision). F32 WMMA has
// no A/B negation modifier, so the complex combine (Xr = RrWr - RiWi,
// Xi = RrWi + RiWr) uses 4 accumulators, resolved once in the epilogue.
// ---------------------------------------------------------------------------
__global__ __launch_bounds__(128) void amb_wmma_kernel(
    const float* __restrict__ s_real,
    const float* __restrict__ s_imag,
    float*       __restrict__ out,
    unsigned int* __restrict__ bmax)
{
    __shared__ float2 lds_s [NFFT];   // signal, complex, 8 KB
    __shared__ float2 lds_tw[NFFT];   // tw[t] = exp(-2*pi*i*t/N), 8 KB
    __shared__ float  red[128];

    const int tid  = threadIdx.x;
    const int wave = tid >> 5;
    const int lane = tid & 31;

    const int b  = blockIdx.x / NKT;                  // batch
    const int t  = blockIdx.x - b * NKT;              // lag tile index 0..32
    const int k0 = t << 4;                            // lag tile base (16 lags)
    const int j0 = (blockIdx.y << 6) + (wave << 4);   // freq tile base (16 freqs)

    // ---- stage signal + twiddle table into LDS (cooperative, 128 threads) ----
    const float* sr = s_real + b * NFFT;
    const float* si = s_imag + b * NFFT;
    const float wstep = -6.28318530717958647692f / (float)NFFT;
    for (int i = tid; i < NFFT; i += 128) {
        lds_s[i] = make_float2(sr[i], si[i]);
        float sv, cv;
        __sincosf(wstep * (float)i, &sv, &cv);
        lds_tw[i] = make_float2(cv, sv);              // (Wr, Wi) of exp(-2pi i t/N)
    }
    __syncthreads();

    // ---- per-lane roles per ISA 16x16x4 f32 operand layout ----
    // A (16x4): lane&15 = row(M=lag), K = 2*(lane>=16) + vgpr
    // B (4x16): lane&15 = col(N=freq), same K mapping
    const int half = lane >> 4;
    const int rc   = lane & 15;
    const int krow = k0 + rc;     // lag for A role
    const int jcol = j0 + rc;     // frequency for B role
    const int kSel = half << 1;

    v8f acc_rr = {0.f, 0.f, 0.f, 0.f, 0.f, 0.f, 0.f, 0.f};  // Rr*Wr
    v8f acc_ii = {0.f, 0.f, 0.f, 0.f, 0.f, 0.f, 0.f, 0.f};  // Ri*Wi
    v8f acc_ri = {0.f, 0.f, 0.f, 0.f, 0.f, 0.f, 0.f, 0.f};  // Rr*Wi
    v8f acc_ir = {0.f, 0.f, 0.f, 0.f, 0.f, 0.f, 0.f, 0.f};  // Ri*Wr

    for (int m0 = 0; m0 < NFFT; m0 += 4) {
        const int mA0 = m0 + kSel;
        const int mA1 = mA0 + 1;

        // A-matrix: R[k, m] = s[m] * conj(s[(m-k) mod N])   (built on the fly)
        const float2 sm0 = lds_s[mA0];
        const float2 sm1 = lds_s[mA1];
        const float2 sd0 = lds_s[(mA0 - krow) & NMASK];
        const float2 sd1 = lds_s[(mA1 - krow) & NMASK];

        v2f Ar, Ai;
        Ar[0] = sm0.x * sd0.x + sm0.y * sd0.y;   // Re(s*conj(d))
        Ai[0] = sm0.y * sd0.x - sm0.x * sd0.y;   // Im(s*conj(d))
        Ar[1] = sm1.x * sd1.x + sm1.y * sd1.y;
        Ai[1] = sm1.y * sd1.x - sm1.x * sd1.y;

        // B-matrix: W[m, j] = tw[(m*j) mod N]
        const float2 w0 = lds_tw[(mA0 * jcol) & NMASK];
        const float2 w1 = lds_tw[(mA1 * jcol) & NMASK];
        v2f Br, Bi;
        Br[0] = w0.x;  Br[1] = w1.x;
        Bi[0] = w0.y;  Bi[1] = w1.y;

        acc_rr = __builtin_amdgcn_wmma_f32_16x16x4_f32(false, Ar, false, Br,
                                                       (short)0, acc_rr, false, false);
        acc_ii = __builtin_amdgcn_wmma_f32_16x16x4_f32(false, Ai, false, Bi,
                                                       (short)0, acc_ii, false, false);
        acc_ri = __builtin_amdgcn_wmma_f32_16x16x4_f32(false, Ar, false, Bi,
                                                       (short)0, acc_ri, false, false);
        acc_ir = __builtin_amdgcn_wmma_f32_16x16x4_f32(false, Ai, false, Br,
                                                       (short)0, acc_ir, false, false);
    }

    // ---- epilogue: chi = |X|^2, fftshift via ^512 (N=1024), symmetric store,
    //      track block max over directly-computed (stored) cells only ----
    // C/D layout: lane L -> col = L&15, row = 8*(L>=16) + vgpr
    float lmax = 0.f;
    const int jj  = jcol ^ 512;                          // fftshifted direct col
    const int jjm = ((NFFT - jcol) & NMASK) ^ 512;       // fftshifted mirrored col
    float* outb = out + (size_t)b * NFFT * NFFT;
#pragma unroll
    for (int v = 0; v < 8; ++v) {
        const int row = (half << 3) + v;
        const int k   = k0 + row;
        const float xr = acc_rr[v] - acc_ii[v];
        const float xi = acc_ri[v] + acc_ir[v];
        const float chi = xr * xr + xi * xi;
        if (k <= 512) {                                  // direct half-plane
            outb[(size_t)(k ^ 512) * NFFT + jj] = chi;
            lmax = fmaxf(lmax, chi);
        }
        if (k >= 1 && k <= 511) {                        // mirrored half-plane
            const int km = NFFT - k;                     // (-k) mod N, in 513..1023
            outb[(size_t)(km ^ 512) * NFFT + jjm] = chi;
        }
    }

    red[tid] = lmax;
    __syncthreads();
    if (tid == 0) {
        float m = 0.f;
        for (int i = 0; i < 128; ++i) m = fmaxf(m, red[i]);
        // chi >= 0, so IEEE float order == unsigned int order on the bit pattern
        atomicMax(bmax + b, __float_as_uint(m));
    }
}

// ---------------------------------------------------------------------------
// Kernel 2: per-batch max normalization, float4 vectorized (512B/wave stores).
// 16 * 1024 * 1024 floats = 2^22 float4; batch = idx >> 18.
// ---------------------------------------------------------------------------
__global__ __launch_bounds__(256) void amb_normalize(
    float4* __restrict__ out, const unsigned int* __restrict__ bmax)
{
    const size_t i = (size_t)blockIdx.x * 256 + threadIdx.x;
    const int b = (int)(i >> 18);                 // 2^18 float4 per batch
    const float inv = 1.0f / __uint_as_float(bmax[b]);
    float4 v = out[i];
    v.x *= inv; v.y *= inv; v.z *= inv; v.w *= inv;
    out[i] = v;
}

// ---------------------------------------------------------------------------
extern "C" void kernel_launch(void* const* d_in, const int* in_sizes, int n_in,
                              void* d_out, int out_size, void* d_ws, size_t ws_size,
                              hipStream_t stream)
{
    const float* s_real = (const float*)d_in[0];
    const float* s_imag = (const float*)d_in[1];
    float* out = (float*)d_out;
    unsigned int* bmax = (unsigned int*)d_ws;

    amb_init_max<<<1, 32, 0, stream>>>(bmax);

    dim3 grid(NBATCH * NKT, NFFT / 64);           // (528, 16)
    amb_wmma_kernel<<<grid, 128, 0, stream>>>(s_real, s_imag, out, bmax);

    const int n4 = NBATCH * NFFT * NFFT / 4;      // 4,194,304 float4
    amb_normalize<<<n4 / 256, 256, 0, stream>>>((float4*)out, bmax);
}